// NoteAttention_60112362274823
// MI455X (gfx1250) — compile-verified
//
#include <hip/hip_runtime.h>

typedef float v2f __attribute__((ext_vector_type(2)));
typedef float v8f __attribute__((ext_vector_type(8)));

#define B_  32
#define S_  2048
#define H_  1024
#define A_  512
#define NCH 16           // sequence chunks for weighted-sum pass
#define SCH (S_ / NCH)   // 128

// ---------------------------------------------------------------------------
// K1: u[b][h] = dot(w[h,:], ctx[b,:])   (tiny: 33 MFLOP, L2-resident)
// ---------------------------------------------------------------------------
__global__ void k_ctx_proj(const float* __restrict__ w,
                           const float* __restrict__ ctx,
                           float* __restrict__ u) {
  int tid = blockIdx.x * blockDim.x + threadIdx.x;  // 0 .. B*H-1
  int b = tid >> 10;
  int h = tid & (H_ - 1);
  const float4* wrow = (const float4*)(w + (size_t)h * A_);
  const float4* crow = (const float4*)(ctx + (size_t)b * A_);
  float s = 0.f;
#pragma unroll 4
  for (int i = 0; i < A_ / 4; ++i) {
    float4 wv = wrow[i], cv = crow[i];
    s += wv.x * cv.x + wv.y * cv.y + wv.z * cv.z + wv.w * cv.w;
  }
  u[tid] = s;
}

// ---------------------------------------------------------------------------
// K2: vu[b,s] = mask[b,s] * dot(x[b,s,:], u[b,:])  — streaming pass 1 over x
// One wave per (b,s) row; coalesced float4 loads; wave32 shuffle reduction.
// ---------------------------------------------------------------------------
__global__ void k_scores(const float* __restrict__ x,
                         const float* __restrict__ u,
                         const float* __restrict__ mask,
                         float* __restrict__ vu) {
  int row  = (blockIdx.x * blockDim.x + threadIdx.x) >> 5;  // (b*S + s)
  int lane = threadIdx.x & 31;
  int b    = row >> 11;  // / S_
  const float4* xr = (const float4*)(x + (size_t)row * H_);
  const float4* ur = (const float4*)(u + (size_t)b * H_);
  float s = 0.f;
#pragma unroll
  for (int i = 0; i < H_ / (32 * 4); ++i) {
    float4 xv = xr[i * 32 + lane];
    float4 uv = ur[i * 32 + lane];
    s += xv.x * uv.x + xv.y * uv.y + xv.z * uv.z + xv.w * uv.w;
  }
#pragma unroll
  for (int off = 16; off > 0; off >>= 1) s += __shfl_xor(s, off, 32);
  if (lane == 0) vu[row] = s * mask[row];
}

// ---------------------------------------------------------------------------
// K3: masked softmax over S per batch row (matches reference semantics:
// subtract row max, exp * mask, divide_no_nan). One block per b.
// ---------------------------------------------------------------------------
__global__ void k_softmax(const float* __restrict__ vu,
                          const float* __restrict__ mask,
                          float* __restrict__ alphas) {
  __shared__ float red[256];
  int b = blockIdx.x, t = threadIdx.x;
  float v[8];
  float m = -3.402823466e38f;
#pragma unroll
  for (int i = 0; i < 8; ++i) {
    v[i] = vu[b * S_ + t + i * 256];
    m = fmaxf(m, v[i]);
  }
  red[t] = m;
  __syncthreads();
  for (int off = 128; off > 0; off >>= 1) {
    if (t < off) red[t] = fmaxf(red[t], red[t + off]);
    __syncthreads();
  }
  m = red[0];
  __syncthreads();
  float e[8], s = 0.f;
#pragma unroll
  for (int i = 0; i < 8; ++i) {
    e[i] = expf(v[i] - m) * mask[b * S_ + t + i * 256];
    s += e[i];
  }
  red[t] = s;
  __syncthreads();
  for (int off = 128; off > 0; off >>= 1) {
    if (t < off) red[t] += red[t + off];
    __syncthreads();
  }
  float denom = red[0];
#pragma unroll
  for (int i = 0; i < 8; ++i)
    alphas[b * S_ + t + i * 256] = (denom == 0.f) ? 0.f : e[i] / denom;
}

// ---------------------------------------------------------------------------
// K4: partial y[b,h] = sum over an S-chunk of alphas[b,s]*x[b,s,h]
// Streaming pass 2 over x; fully coalesced; deterministic (no atomics).
// ---------------------------------------------------------------------------
__global__ void k_wsum_part(const float* __restrict__ x,
                            const float* __restrict__ alphas,
                            float* __restrict__ yp) {
  int b  = blockIdx.x >> 4;   // / NCH
  int ch = blockIdx.x & (NCH - 1);
  int t  = threadIdx.x;       // 0..255 -> float4 column index
  float4 acc = make_float4(0.f, 0.f, 0.f, 0.f);
  const float* xb = x + ((size_t)b * S_ + (size_t)ch * SCH) * H_;
  const float* ab = alphas + (size_t)b * S_ + ch * SCH;
  for (int s = 0; s < SCH; ++s) {
    float a = ab[s];
    float4 xv = ((const float4*)(xb + (size_t)s * H_))[t];
    acc.x += a * xv.x;
    acc.y += a * xv.y;
    acc.z += a * xv.z;
    acc.w += a * xv.w;
  }
  ((float4*)(yp + ((size_t)ch * B_ + b) * H_))[t] = acc;
}

// K4b: y = sum of NCH partials
__global__ void k_wsum_reduce(const float* __restrict__ yp,
                              float* __restrict__ y) {
  int t = blockIdx.x * blockDim.x + threadIdx.x;  // float4 index over B*H/4
  float4 acc = make_float4(0.f, 0.f, 0.f, 0.f);
#pragma unroll
  for (int c = 0; c < NCH; ++c) {
    float4 v = ((const float4*)(yp + (size_t)c * B_ * H_))[t];
    acc.x += v.x; acc.y += v.y; acc.z += v.z; acc.w += v.w;
  }
  ((float4*)y)[t] = acc;
}

// ---------------------------------------------------------------------------
// K5: output(32x512) = y(32x1024) @ w(1024x512), fp32 WMMA 16x16x4.
// One wave per 16x16 tile. A-frag: lane l<16 row M=l holds K=k,k+1; lanes
// 16..31 hold K=k+2,k+3. B-frag: VGPR0 = rows K=k / k+2 striped across lane
// halves, VGPR1 = rows K=k+1 / k+3 (same pattern as C/D row striping).
// ---------------------------------------------------------------------------
__global__ void k_out_gemm(const float* __restrict__ y,
                           const float* __restrict__ w,
                           float* __restrict__ out) {
  int lane = threadIdx.x & 31;
  int half = lane >> 4;
  int l    = lane & 15;
  int tM   = blockIdx.x & 1;   // 2 row tiles (B=32)
  int tN   = blockIdx.x >> 1;  // 32 col tiles (A=512)

  const float* Arow = y + (size_t)(tM * 16 + l) * H_;
  const float* Bcol = w + tN * 16 + l;

  v8f c = {};
  for (int k = 0; k < H_; k += 4) {
    v2f a, bb;
    a.x  = Arow[k + 2 * half];
    a.y  = Arow[k + 2 * half + 1];
    bb.x = Bcol[(size_t)(k + 2 * half) * A_];
    bb.y = Bcol[(size_t)(k + 2 * half + 1) * A_];
    c = __builtin_amdgcn_wmma_f32_16x16x4_f32(
        /*neg_a=*/false, a, /*neg_b=*/false, bb,
        /*c_mod=*/(short)0, c, /*reuse_a=*/false, /*reuse_b=*/false);
  }
#pragma unroll
  for (int v = 0; v < 8; ++v) {
    int row = tM * 16 + v + half * 8;
    out[(size_t)row * A_ + tN * 16 + l] = c[v];
  }
}

// ---------------------------------------------------------------------------
extern "C" void kernel_launch(void* const* d_in, const int* in_sizes, int n_in,
                              void* d_out, int out_size, void* d_ws, size_t ws_size,
                              hipStream_t stream) {
  const float* x    = (const float*)d_in[0];  // (B,S,H)
  const float* ctx  = (const float*)d_in[1];  // (B,A)
  const float* mask = (const float*)d_in[2];  // (B,S)
  const float* w    = (const float*)d_in[3];  // (H,A)

  float* out    = (float*)d_out;        // (B,A) = 16384 floats
  float* alphas = out + B_ * A_;        // (B,S) = 65536 floats

  float* u  = (float*)d_ws;             // B*H   = 32768
  float* vu = u + B_ * H_;              // B*S   = 65536
  float* y  = vu + B_ * S_;             // B*H   = 32768
  float* yp = y + B_ * H_;              // NCH*B*H = 524288  (~2.6 MB total ws)

  k_ctx_proj   <<<(B_ * H_) / 256, 256, 0, stream>>>(w, ctx, u);
  k_scores     <<<(B_ * S_ * 32) / 256, 256, 0, stream>>>(x, u, mask, vu);
  k_softmax    <<<B_, 256, 0, stream>>>(vu, mask, alphas);
  k_wsum_part  <<<B_ * NCH, 256, 0, stream>>>(x, alphas, yp);
  k_wsum_reduce<<<(B_ * H_ / 4) / 256, 256, 0, stream>>>(yp, y);
  k_out_gemm   <<<(B_ / 16) * (A_ / 16), 32, 0, stream>>>(y, w, out);
}